// DiffGFDNSinglePos_38482906972590
// MI455X (gfx1250) — compile-verified
//
#include <hip/hip_runtime.h>
#include <math.h>

typedef float v2f __attribute__((ext_vector_type(2)));
typedef float v8f __attribute__((ext_vector_type(8)));

#define NLINES 32
#define KPTS   32768
#define NSEC   8              // LP_ORDER/2 biquad sections
#define WAVES_PER_BLOCK 8
#define FREQ_PER_WAVE   16
#define ROWSTRIDE 33          // 32 rows + 1 pad -> conflict-free LDS
#define PI_F 3.14159265358979f

// ---------------------------------------------------------------------------
// Kernel A: build AG = feedback_matrix * diag(gamma), gamma = g^delay, and
// derive the Neumann iteration count J from the spectral bound max(gamma).
// ---------------------------------------------------------------------------
__global__ void gfdn_setup(const float* __restrict__ F,
                           const float* __restrict__ g,
                           const int*   __restrict__ delays,
                           float* __restrict__ AG, int* __restrict__ Jout)
{
    __shared__ float gamma[NLINES];
    int t = threadIdx.x;
    if (t < NLINES) {
        gamma[t] = (float)pow((double)g[t], (double)delays[t]);
    }
    __syncthreads();
    for (int i = t; i < NLINES * NLINES; i += blockDim.x) {
        AG[i] = F[i] * gamma[i & 31];          // column scaling: A @ diag(gamma)
    }
    if (t == 0) {
        float gmax = 0.f;
        for (int i = 0; i < NLINES; ++i) gmax = fmaxf(gmax, fabsf(gamma[i]));
        gmax = fminf(fmaxf(gmax, 1e-3f), 0.9995f);
        int J = (int)ceilf(logf(1e-8f) / logf(gmax));
        J = min(max(J, 8), 512);
        *Jout = J;
    }
}

__device__ __forceinline__ float swap16(float x) {
    // ds_swizzle group-of-32, xor_mask=0x10, and_mask=0x1f  (SWAPX16)
    return __int_as_float(__builtin_amdgcn_ds_swizzle(__float_as_int(x), 0x401f));
}

// ---------------------------------------------------------------------------
// Kernel B: per wave, 16 frequencies. Neumann resolvent with f32 WMMA GEMMs.
// ---------------------------------------------------------------------------
__launch_bounds__(WAVES_PER_BLOCK * 32)
__global__ void gfdn_freq_response(const float* __restrict__ AG,
                                   const int*   __restrict__ Jptr,
                                   const float* __restrict__ bgain,   // [N,1]
                                   const float* __restrict__ cgain,   // [N,1]
                                   const int*   __restrict__ delays,  // [N]
                                   const float* __restrict__ direct_r,
                                   const float* __restrict__ direct_i,
                                   const float* __restrict__ sos,     // [8][6]
                                   float* __restrict__ out)           // [K] cplx
{
    __shared__ float Ush[WAVES_PER_BLOCK][2][FREQ_PER_WAVE][ROWSTRIDE];

    const int lane = threadIdx.x & 31;
    const int wave = threadIdx.x >> 5;
    const int col  = lane & 15;     // freq within tile / N-column
    const int hi   = lane >> 4;     // which half-lane group
    const int kf   = (blockIdx.x * WAVES_PER_BLOCK + wave) * FREQ_PER_WAVE + col;

    float (*U)[FREQ_PER_WAVE][ROWSTRIDE] = Ush[wave];   // [re/im][col][row]

    // ---- A fragments: AG held in registers for the whole solve -----------
    // 16x4 f32 A layout: lanes 0-15 M=0..15 K={0,1}; lanes 16-31 K={2,3}.
    v2f aF[2][8];
    #pragma unroll
    for (int mi = 0; mi < 2; ++mi)
        #pragma unroll
        for (int kq = 0; kq < 8; ++kq) {
            const int r  = mi * 16 + col;
            const int k0 = kq * 4 + 2 * hi;
            aF[mi][kq].x = AG[r * 32 + k0];
            aF[mi][kq].y = AG[r * 32 + k0 + 1];
        }

    // ---- per-lane row constants: w = z^{-m} (exact integer phase), c, b --
    float wr[16], wi[16], cw[16];
    float accR = 0.f, accI = 0.f;
    #pragma unroll
    for (int mi = 0; mi < 2; ++mi)
        #pragma unroll
        for (int v = 0; v < 8; ++v) {
            const int row = mi * 16 + 8 * hi + v;
            const int idx = mi * 8 + v;
            const unsigned p = ((unsigned)kf * (unsigned)delays[row]) & (2u * KPTS - 1u);
            float s, c;
            __sincosf(-PI_F * ((float)p / (float)KPTS), &s, &c);
            wr[idx] = c; wi[idx] = s;
            cw[idx] = cgain[row];
            const float b  = bgain[row];
            const float ur = b * c, ui = b * s;          // u0 = D^-1 b
            accR += cw[idx] * ur;  accI += cw[idx] * ui; // H += c^T u0
            U[0][col][row] = ur;   U[1][col][row] = ui;
        }
    asm volatile("s_wait_dscnt 0" ::: "memory");

    const int J = *Jptr;
    for (int j = 0; j < J; ++j) {
        // ---- B fragments from LDS (4x16 tiles: rows kq*4+2*hi, +1) -------
        v2f bF[2][8];
        #pragma unroll
        for (int nb = 0; nb < 2; ++nb)
            #pragma unroll
            for (int kq = 0; kq < 8; ++kq) {
                const float* p = &U[nb][col][kq * 4 + 2 * hi];
                bF[nb][kq].x = p[0];
                bF[nb][kq].y = p[1];
            }

        // ---- t = AG * u  : 32 x V_WMMA_F32_16X16X4_F32 -------------------
        v8f t00 = {}, t01 = {}, t10 = {}, t11 = {};
        #pragma unroll
        for (int kq = 0; kq < 8; ++kq) {
            t00 = __builtin_amdgcn_wmma_f32_16x16x4_f32(false, aF[0][kq], false, bF[0][kq], (short)0, t00, false, false);
            t01 = __builtin_amdgcn_wmma_f32_16x16x4_f32(false, aF[0][kq], false, bF[1][kq], (short)0, t01, false, false);
            t10 = __builtin_amdgcn_wmma_f32_16x16x4_f32(false, aF[1][kq], false, bF[0][kq], (short)0, t10, false, false);
            t11 = __builtin_amdgcn_wmma_f32_16x16x4_f32(false, aF[1][kq], false, bF[1][kq], (short)0, t11, false, false);
        }
        asm volatile("s_wait_dscnt 0" ::: "memory");   // all B reads retired

        // ---- u = w ∘ t, accumulate H, write back to LDS ------------------
        #pragma unroll
        for (int mi = 0; mi < 2; ++mi) {
            const v8f tr8 = mi ? t10 : t00;
            const v8f ti8 = mi ? t11 : t01;
            #pragma unroll
            for (int v = 0; v < 8; ++v) {
                const int row = mi * 16 + 8 * hi + v;
                const int idx = mi * 8 + v;
                const float tr = tr8[v], ti = ti8[v];
                const float ur = wr[idx] * tr - wi[idx] * ti;
                const float ui = wr[idx] * ti + wi[idx] * tr;
                accR += cw[idx] * ur;  accI += cw[idx] * ui;
                U[0][col][row] = ur;   U[1][col][row] = ui;
            }
        }
        asm volatile("s_wait_dscnt 0" ::: "memory");
    }

    // ---- combine lane halves, apply direct path + biquad cascade ---------
    float hR = accR + swap16(accR);
    float hI = accI + swap16(accI);
    if (hi == 0) {
        hR += direct_r[kf];
        hI += direct_i[kf];
        // zinv = exp(-i*pi*f)
        float s, c;
        __sincosf(-PI_F * ((float)kf / (float)KPTS), &s, &c);
        const float z2r = c * c - s * s, z2i = 2.f * c * s;
        float rr = 1.f, ri = 0.f;
        #pragma unroll
        for (int sc = 0; sc < NSEC; ++sc) {
            const float* sp = sos + sc * 6;
            const float b0 = sp[0], b1 = sp[1], b2 = sp[2];
            const float a0 = sp[3], a1 = sp[4], a2 = sp[5];
            const float nr = b0 + b1 * c + b2 * z2r;
            const float ni =      b1 * s + b2 * z2i;
            const float dr = a0 + a1 * c + a2 * z2r;
            const float di =      a1 * s + a2 * z2i;
            const float inv = 1.f / (dr * dr + di * di);
            const float qr = (nr * dr + ni * di) * inv;
            const float qi = (ni * dr - nr * di) * inv;
            const float t  = rr * qr - ri * qi;
            ri = rr * qi + ri * qr;
            rr = t;
        }
        out[2 * kf]     = hR * rr - hI * ri;
        out[2 * kf + 1] = hR * ri + hI * rr;
    }
}

// ---------------------------------------------------------------------------
extern "C" void kernel_launch(void* const* d_in, const int* in_sizes, int n_in,
                              void* d_out, int out_size, void* d_ws, size_t ws_size,
                              hipStream_t stream)
{
    // setup_inputs() order:
    // 0 freqs[K] 1 input_gains[N,1] 2 output_gains[N,1] 3 feedback_matrix[N,N]
    // 4 gain_per_sample[N] 5 direct_real[K] 6 direct_imag[K] 7 sos[8,6] 8 delays[N]
    const float* bgain   = (const float*)d_in[1];
    const float* cgain   = (const float*)d_in[2];
    const float* fbmat   = (const float*)d_in[3];
    const float* gps     = (const float*)d_in[4];
    const float* dre     = (const float*)d_in[5];
    const float* dim     = (const float*)d_in[6];
    const float* sos     = (const float*)d_in[7];
    const int*   delays  = (const int*)d_in[8];

    float* AG   = (float*)d_ws;                 // 1024 floats
    int*   Jptr = (int*)((float*)d_ws + 1024);  // 1 int

    gfdn_setup<<<1, 256, 0, stream>>>(fbmat, gps, delays, AG, Jptr);

    const int grid = KPTS / (WAVES_PER_BLOCK * FREQ_PER_WAVE);  // 256 blocks
    gfdn_freq_response<<<grid, WAVES_PER_BLOCK * 32, 0, stream>>>(
        AG, Jptr, bgain, cgain, delays, dre, dim, sos, (float*)d_out);
}